// TopKPooling_71597104824916
// MI455X (gfx1250) — compile-verified
//
#include <hip/hip_runtime.h>
#include <math.h>

#define BB 32
#define NN 2048
#define FF 128
#define KK 512

typedef float v2f __attribute__((ext_vector_type(2)));
typedef float v8f __attribute__((ext_vector_type(8)));
typedef unsigned int u32x4 __attribute__((ext_vector_type(4)));
typedef int i32x4 __attribute__((ext_vector_type(4)));
typedef int i32x8 __attribute__((ext_vector_type(8)));

#if defined(__has_builtin)
#if __has_builtin(__builtin_amdgcn_tensor_load_to_lds)
#define HAVE_TDM 1
#endif
#endif

#ifdef HAVE_TDM
#if __has_include(<hip/amd_detail/amd_gfx1250_TDM.h>)
#pragma message("CDNA5-DIAG: TDM builtin available, 6-arg (therock header) form")
#define TDM_SIX_ARG 1
#else
#pragma message("CDNA5-DIAG: TDM builtin available, 5-arg (ROCm 7.2) form")
#endif
#else
#pragma message("CDNA5-DIAG: TDM builtin NOT available, using VALU fallback gather")
#endif

// ---------------------------------------------------------------------------
// Kernel 1: scores[b,n] = dot(H[b,n,:], w)  via V_WMMA_F32_16X16X4_F32.
// A tile = 16 nodes x 4 features (f32); B tile = 4 features broadcast over 16
// cols of w. Column 0 of D then holds the 16 node scores (all columns equal).
// Also computes ||w|| once into *normp.
// ---------------------------------------------------------------------------
__global__ __launch_bounds__(256) void tk_scores(const float* __restrict__ H,
                                                 const float* __restrict__ w,
                                                 float* __restrict__ scores,
                                                 float* __restrict__ normp) {
  const int b = blockIdx.x;
  const int tid = threadIdx.x;
  const int wave = tid >> 5;   // 0..7
  const int lane = tid & 31;
  const int m = lane & 15;     // A-matrix row M (16x4 f32 layout)
  const int hi = lane >> 4;    // 0: K=0,1  1: K=2,3

  if (b == 0 && tid == 0) {
    float ss = 0.f;
    for (int f = 0; f < FF; ++f) ss += w[f] * w[f];
    *normp = sqrtf(ss);
  }

  const float* Hb = H + (size_t)b * NN * FF;

  for (int t = 0; t < 16; ++t) {
    const int base = wave * 256 + t * 16;   // 16-node tile
    v8f c = {};
    const float* arow = Hb + (size_t)(base + m) * FF + 2 * hi;
    const float* brow = w + 2 * hi;
#pragma unroll 4
    for (int k0 = 0; k0 < FF; k0 += 4) {
      v2f a = *(const v2f*)(arow + k0);    // A[M][k0+2hi .. +1]
      v2f bf = *(const v2f*)(brow + k0);   // B[k0+2hi .. +1][*] broadcast
      c = __builtin_amdgcn_wmma_f32_16x16x4_f32(
          false, a, false, bf, (short)0, c, false, false);
    }
    // C/D layout: VGPR v holds M=v (lanes 0-15) / M=v+8 (lanes 16-31), N=lane&15.
    // Take column N==0 (lanes 0 and 16): 16 scores per tile.
    if (m == 0) {
      float* srow = scores + (size_t)b * NN + base + hi * 8;
#pragma unroll
      for (int v = 0; v < 8; ++v) srow[v] = c[v];
    }
  }
}

// ---------------------------------------------------------------------------
// Kernel 2: per-batch bitonic top-K. Sort 2048 (score,idx) pairs descending by
// score (ties: ascending idx, matching jax.lax.top_k / torch.topk), keep 512.
// ---------------------------------------------------------------------------
__global__ __launch_bounds__(1024) void tk_topk(const float* __restrict__ scores,
                                                int* __restrict__ topk_idx,
                                                float* __restrict__ topk_sc) {
  __shared__ float s[NN];
  __shared__ int si[NN];
  const int b = blockIdx.x;
  const int tid = threadIdx.x;

  for (int i = tid; i < NN; i += 1024) {
    s[i] = scores[(size_t)b * NN + i];
    si[i] = i;
  }
  __syncthreads();

  for (int k = 2; k <= NN; k <<= 1) {
    for (int j = k >> 1; j > 0; j >>= 1) {
      for (int i = tid; i < NN; i += 1024) {
        const int l = i ^ j;
        if (l > i) {
          const bool best_first = ((i & k) == 0);
          float sa = s[i], sb = s[l];
          int ia = si[i], ib = si[l];
          const bool a_first = (sa > sb) || (sa == sb && ia < ib);
          if (best_first != a_first) {
            s[i] = sb; s[l] = sa;
            si[i] = ib; si[l] = ia;
          }
        }
      }
      __syncthreads();
    }
  }
  if (tid < KK) {
    topk_idx[(size_t)b * KK + tid] = si[tid];
    topk_sc[(size_t)b * KK + tid] = s[tid];
  }
}

// ---------------------------------------------------------------------------
// Kernel 3: H_retained[b,k,:] = H[b, idx[b,k], :] * sigmoid(score/norm)
// 128 threads handle 4 retained rows per block with float4 (b128) traffic.
// ---------------------------------------------------------------------------
__global__ __launch_bounds__(128) void tk_hgather(const float* __restrict__ H,
                                                  const int* __restrict__ topk_idx,
                                                  const float* __restrict__ topk_sc,
                                                  const float* __restrict__ normp,
                                                  float* __restrict__ out0) {
  const int b = blockIdx.y;
  const int k0 = blockIdx.x * 4;               // 4 rows per block
  const int tid = threadIdx.x;
  const int r = tid >> 5;                      // row within block (0..3)
  const int f4 = (tid & 31);                   // float4 slot (32 * 4 = 128 feats)
  const float inv_norm = 1.0f / *normp;

  const size_t bk = (size_t)b * KK + k0 + r;
  const int idx = topk_idx[bk];
  const float sc = topk_sc[bk] * inv_norm;
  const float gate = 1.0f / (1.0f + __expf(-sc));

  const float4 hv =
      *(const float4*)(H + ((size_t)b * NN + idx) * FF + f4 * 4);
  float4 o;
  o.x = hv.x * gate; o.y = hv.y * gate; o.z = hv.z * gate; o.w = hv.w * gate;
  *(float4*)(out0 + bk * FF + f4 * 4) = o;
}

// ---------------------------------------------------------------------------
// Kernel 4: G_retained[b,i,j] = G[b, idx[b,i], idx[b,j]].
// Per block: TDM gather-mode TENSOR_LOAD_TO_LDS streams 8 index-selected rows
// (8 x 2048 f32 = 64KB) of G[b] into LDS (DMA overlapped with the column-index
// copy); then threads pick the 512 selected columns from LDS and write
// coalesced output rows.
// ---------------------------------------------------------------------------
#define ROWS_PER_BLK 8

__global__ __launch_bounds__(256) void tk_ggather(const float* __restrict__ G,
                                                  const int* __restrict__ topk_idx,
                                                  float* __restrict__ out1) {
  extern __shared__ float rows[];          // ROWS_PER_BLK * NN floats
  __shared__ int sidx[KK];                 // column indices

  const int b = blockIdx.y;
  const int i0 = blockIdx.x * ROWS_PER_BLK;
  const int tid = threadIdx.x;
  const int* tib = topk_idx + (size_t)b * KK;

#ifdef HAVE_TDM
  if (tid < 32) {
    const unsigned long long ga =
        (unsigned long long)(const void*)(G + (size_t)b * NN * NN);
    const unsigned lds_off = (unsigned)(unsigned long)(void*)rows;

    // D# group 0: count=1, gather_mode=1, 16-bit indices, type=2 ("image")
    u32x4 g0;
    g0[0] = 0x80000001u;                       // gather_mode | count=1
    g0[1] = lds_off;                           // lds_addr (bytes)
    g0[2] = (unsigned)(ga & 0xFFFFFFFFu);      // global_addr[31:0]
    g0[3] = (unsigned)((ga >> 32) & 0x01FFFFFFu) | 0x80000000u; // [56:32]|type=2

    // D# group 1
    i32x8 g1;
    g1[0] = 0x00020000;                  // workgroup_mask=0, data_size=2 (4B)
    g1[1] = (int)((NN & 0xFFFF) << 16);  // tensor_dim0[15:0] @ bits 63:48
    g1[2] = (int)((NN & 0xFFFF) << 16);  // dim0 hi=0; tensor_dim1[15:0] @ 95:80
    g1[3] = (int)((NN & 0xFFFFu) << 16); // dim1 hi=0; tile_dim0=2048 @ 127:112
    g1[4] = ROWS_PER_BLK;                // tile_dim1 = #valid gather indices
    g1[5] = NN;                          // tensor_dim0_stride[31:0]
    g1[6] = 0;
    g1[7] = 0;

    // D# groups 2/3: 16-bit row indices (only first 8 valid)
    const int* ri = tib + i0;
    i32x4 g2, g3;
    g2[0] = (ri[0] & 0xFFFF) | (ri[1] << 16);
    g2[1] = (ri[2] & 0xFFFF) | (ri[3] << 16);
    g2[2] = (ri[4] & 0xFFFF) | (ri[5] << 16);
    g2[3] = (ri[6] & 0xFFFF) | (ri[7] << 16);
    g3[0] = 0; g3[1] = 0; g3[2] = 0; g3[3] = 0;

#ifdef TDM_SIX_ARG
    i32x8 gx = {0, 0, 0, 0, 0, 0, 0, 0};
    __builtin_amdgcn_tensor_load_to_lds(g0, g1, g2, g3, gx, 0);
#else
    __builtin_amdgcn_tensor_load_to_lds(g0, g1, g2, g3, 0);
#endif
  }
  // Overlap the 64KB row DMA with the 2KB column-index fill.
  for (int j = tid; j < KK; j += 256) sidx[j] = tib[j];
  if (tid < 32) {
    __builtin_amdgcn_s_wait_tensorcnt(0);
  }
#else
  for (int j = tid; j < KK; j += 256) sidx[j] = tib[j];
  // Fallback: coalesced VALU copy of the 8 selected rows into LDS.
  for (int e = tid; e < ROWS_PER_BLK * NN; e += 256) {
    const int r = e >> 11;
    const int col = e & (NN - 1);
    rows[e] = G[((size_t)b * NN + tib[i0 + r]) * NN + col];
  }
#endif
  __syncthreads();

  float* outb = out1 + ((size_t)b * KK + i0) * KK;
  for (int e = tid; e < ROWS_PER_BLK * KK; e += 256) {
    const int r = e >> 9;          // row in chunk
    const int j = e & (KK - 1);    // output column
    outb[(size_t)r * KK + j] = rows[r * NN + sidx[j]];
  }
}

// ---------------------------------------------------------------------------
extern "C" void kernel_launch(void* const* d_in, const int* in_sizes, int n_in,
                              void* d_out, int out_size, void* d_ws, size_t ws_size,
                              hipStream_t stream) {
  const float* H = (const float*)d_in[0];            // [B,N,F]
  const float* G = (const float*)d_in[1];            // [B,N,N]
  const float* w = (const float*)d_in[2];            // [F,1]

  float* out0 = (float*)d_out;                       // [B,K,F]
  float* out1 = out0 + (size_t)BB * KK * FF;         // [B,K,K]

  float* scores = (float*)d_ws;                      // B*N
  float* normp = scores + (size_t)BB * NN;           // 1
  int* topk_idx = (int*)(normp + 1);                 // B*K
  float* topk_sc = (float*)(topk_idx + (size_t)BB * KK); // B*K

  tk_scores<<<BB, 256, 0, stream>>>(H, w, scores, normp);
  tk_topk<<<BB, 1024, 0, stream>>>(scores, topk_idx, topk_sc);
  tk_hgather<<<dim3(KK / 4, BB), 128, 0, stream>>>(H, topk_idx, topk_sc, normp,
                                                   out0);
  tk_ggather<<<dim3(KK / ROWS_PER_BLK, BB), 256,
               ROWS_PER_BLK * NN * sizeof(float), stream>>>(G, topk_idx, out1);
}